// TreeAggregation_83296595739250
// MI455X (gfx1250) — compile-verified
//
#include <hip/hip_runtime.h>
#include <math.h>
#include <stdint.h>

#define DIM 256            // embedding dimension
#define RPC 16             // rows per LDS tile
#define THREADS 256        // 8 wave32s

// ---------------------------------------------------------------------------
// Kernel 1: exclusive prefix-sum of tree_sizes -> offs[0..n], offs[n] = total.
// Single block, 256 threads, each owning a contiguous chunk.
// ---------------------------------------------------------------------------
__global__ void tree_scan_offsets(const int* __restrict__ sizes,
                                  int* __restrict__ offs, int n) {
    __shared__ int partial[256];
    const int tid = threadIdx.x;
    const int per = (n + 255) >> 8;
    const int begin = tid * per;

    int s = 0;
    for (int i = 0; i < per; ++i) {
        const int j = begin + i;
        if (j < n) s += sizes[j];
    }
    partial[tid] = s;
    __syncthreads();

    // Hillis-Steele inclusive scan over the 256 per-thread partials.
    for (int off = 1; off < 256; off <<= 1) {
        const int add = (tid >= off) ? partial[tid - off] : 0;
        __syncthreads();
        partial[tid] += add;
        __syncthreads();
    }

    int run = (tid == 0) ? 0 : partial[tid - 1];   // exclusive base
    for (int i = 0; i < per; ++i) {
        const int j = begin + i;
        if (j < n) { offs[j] = run; run += sizes[j]; }
    }
    if (tid == 0) offs[n] = partial[255];
}

// ---------------------------------------------------------------------------
// Kernel 2: one block per tree. Stream the tree's rows through double-buffered
// LDS tiles using CDNA5 async global->LDS loads; each thread keeps the running
// max of its column in a register.
// ---------------------------------------------------------------------------
__global__ __launch_bounds__(THREADS)
void tree_segmax(const float* __restrict__ emb,
                 const int* __restrict__ offs,
                 float* __restrict__ out) {
    __shared__ float buf[2][RPC * DIM];   // 2 x 16 KB

    const int t   = blockIdx.x;
    const int tid = threadIdx.x;
    const int start = offs[t];
    const int n     = offs[t + 1] - start;      // >= 1 by construction
    const float* base = emb + (size_t)start * DIM;
    const int nchunks = (n + RPC - 1) / RPC;

    // Issue one 16-row tile (16 KB) into buffer p via async loads.
    // 1024 x 16B chunks; 4 per thread. Rows beyond n-1 clamp to n-1
    // (max is idempotent, so duplicates are harmless).
    auto issue = [&](int c, int p) {
        const unsigned lds_base =
            (unsigned)(uintptr_t)(&buf[p][0]);   // LDS offset = addr[31:0]
#pragma unroll
        for (int k = 0; k < 4; ++k) {
            const int q = tid + k * THREADS;     // 16B-chunk id in tile
            int row = c * RPC + (q >> 6);        // 64 chunks per row
            if (row >= n) row = n - 1;
            const unsigned goff =
                (unsigned)row * (DIM * 4u) + (unsigned)(q & 63) * 16u;
            const unsigned lds = lds_base + (unsigned)q * 16u;
            asm volatile("global_load_async_to_lds_b128 %0, %1, %2"
                         :: "v"(lds), "v"(goff), "s"(base)
                         : "memory");
        }
    };

    float m = -INFINITY;

    issue(0, 0);
    for (int c = 0; c < nchunks; ++c) {
        const int p = c & 1;
        if (c + 1 < nchunks) {
            issue(c + 1, p ^ 1);
            // Async loads complete in order: <=4 outstanding means tile c done.
            asm volatile("s_wait_asynccnt 4" ::: "memory");
        } else {
            asm volatile("s_wait_asynccnt 0" ::: "memory");
        }
        __syncthreads();                  // tile c visible to all waves
#pragma unroll
        for (int r = 0; r < RPC; ++r)
            m = fmaxf(m, buf[p][r * DIM + tid]);   // bank-conflict-free
        __syncthreads();                  // done reading before tile reuse
    }

    out[(size_t)t * DIM + tid] = m;
}

// ---------------------------------------------------------------------------
// Launch wrapper
// ---------------------------------------------------------------------------
extern "C" void kernel_launch(void* const* d_in, const int* in_sizes, int n_in,
                              void* d_out, int out_size, void* d_ws, size_t ws_size,
                              hipStream_t stream) {
    (void)n_in; (void)out_size; (void)ws_size;

    const float* emb   = (const float*)d_in[0];
    const int*   sizes = (const int*)d_in[1];     // int32 per harness convention
    float*       out   = (float*)d_out;
    const int num_trees = in_sizes[1];

    int* offs = (int*)d_ws;                       // num_trees + 1 int32s

    tree_scan_offsets<<<1, 256, 0, stream>>>(sizes, offs, num_trees);
    tree_segmax<<<num_trees, THREADS, 0, stream>>>(emb, offs, out);
}